// Transformer_attn_80951543595319
// MI455X (gfx1250) — compile-verified
//
#include <hip/hip_runtime.h>
#include <math.h>

typedef float v2f __attribute__((ext_vector_type(2)));
typedef float v8f __attribute__((ext_vector_type(8)));

#define BATCH 128
#define CCH   48
#define HH    64
#define WW    64
#define PP    32
#define HW    (HH*WW)        // 4096
#define CHW   (CCH*HW)       // 196608
#define OUT_IMG ((size_t)BATCH*CHW)  // 25165824 floats

// ---------------------------------------------------------------------------
// Kernel 1: G[c,c'] = sum_o Wq1[o,c]Wk1[o,c'] + Wq2[o,c]Wk2[o,c'] + Wq3[o,c]Wk3[o,c']
// ---------------------------------------------------------------------------
__global__ void k_build_G(const float* __restrict__ Wq1, const float* __restrict__ Wq2,
                          const float* __restrict__ Wq3, const float* __restrict__ Wk1,
                          const float* __restrict__ Wk2, const float* __restrict__ Wk3,
                          float* __restrict__ G) {
  int idx = blockIdx.x * blockDim.x + threadIdx.x;
  if (idx >= CCH * CCH) return;
  int c = idx / CCH, cp = idx % CCH;
  float acc = 0.f;
  for (int o = 0; o < CCH; ++o) {
    acc += Wq1[o*CCH + c] * Wk1[o*CCH + cp];
    acc += Wq2[o*CCH + c] * Wk2[o*CCH + cp];
    acc += Wq3[o*CCH + c] * Wk3[o*CCH + cp];
  }
  G[idx] = acc;
}

// ---------------------------------------------------------------------------
// Kernel 2: one workgroup per (batch b, parity pi). 128 threads = 4 wave32.
// Wave w computes Gram C_nm = X_n^T X_m  (n = pi+2*(w&1), m = pi+2*(w>>1))
// via v_wmma_f32_16x16x4_f32, contracts with G, builds attn/softmax/det,
// then does the elementwise output pass (coalesced along j).
// ---------------------------------------------------------------------------
__device__ __forceinline__ int patch_off(int p) {
  return ((p >> 1) * PP) * WW + (p & 1) * PP;
}

__global__ __launch_bounds__(128)
void k_main(const float* __restrict__ x, const float* __restrict__ G,
            const float* __restrict__ off_p, const float* __restrict__ off2_p,
            const float* __restrict__ off3_p,
            float* __restrict__ out, float* __restrict__ ld_partial) {
  __shared__ float sG[CCH * CCH];
  __shared__ float sS[4];

  const int tid  = threadIdx.x;
  const int gid  = blockIdx.x;
  const int b    = gid >> 1;
  const int pi   = gid & 1;

  for (int i = tid; i < CCH * CCH; i += 128) sG[i] = G[i];
  __syncthreads();

  const int wave   = tid >> 5;
  const int lane   = tid & 31;
  const int lane15 = lane & 15;
  const int half   = lane >> 4;

  const int rn = wave & 1;          // score row index (patch pi + 2*rn)
  const int rm = (wave >> 1) & 1;   // score col index (patch pi + 2*rm)
  const int n  = pi + 2 * rn;
  const int m  = pi + 2 * rm;

  const float* xb = x + (size_t)b * CHW;
  const int offN = patch_off(n);
  const int offM = patch_off(m);

  // 3x3 accumulator tiles: C_nm[16*tr + row, 16*tc + col], fp32
  v8f acc[3][3];
#pragma unroll
  for (int a = 0; a < 3; ++a)
#pragma unroll
    for (int c2 = 0; c2 < 3; ++c2) acc[a][c2] = {};

  // K = 512 pixels (i in [0,32), jj in [0,16), j = 2*jj + pi), 4 per WMMA step
  for (int k = 0; k < 128; ++k) {
    const int i   = k >> 2;
    const int jj0 = (k & 3) * 4;
    const int jj  = jj0 + 2 * half;       // lane-half selects pix {0,1} vs {2,3}
    const int j0  = 2 * jj + pi;          // vgpr0 pixel column; vgpr1 at j0+2
    const int rcN = offN + i * WW + j0;
    const int rcM = offM + i * WW + j0;

    v2f aF[3], bF[3];
#pragma unroll
    for (int t = 0; t < 3; ++t) {
      const int c = 16 * t + lane15;      // A rows / B cols = channels
      const float* pn = xb + (size_t)c * HW + rcN;
      const float* pm = xb + (size_t)c * HW + rcM;
      aF[t].x = pn[0]; aF[t].y = pn[2];   // A: 16(c) x 4(pix), X_n^T fragment
      bF[t].x = pm[0]; bF[t].y = pm[2];   // B: 4(pix) x 16(c'), X_m fragment
    }
#pragma unroll
    for (int tr = 0; tr < 3; ++tr)
#pragma unroll
      for (int tc = 0; tc < 3; ++tc)
        acc[tr][tc] = __builtin_amdgcn_wmma_f32_16x16x4_f32(
            false, aF[tr], false, bF[tc], (short)0, acc[tr][tc], false, false);
  }

  // S[n,m] = <G, C_nm>_F  using the C/D tile layout:
  // vgpr v, lane l -> row = 16*tr + v + 8*(l>=16), col = 16*tc + (l&15)
  float partial = 0.f;
#pragma unroll
  for (int tr = 0; tr < 3; ++tr) {
#pragma unroll
    for (int tc = 0; tc < 3; ++tc) {
      const int cp = 16 * tc + lane15;
#pragma unroll
      for (int v = 0; v < 8; ++v) {
        const int c = 16 * tr + v + 8 * half;
        partial += acc[tr][tc][v] * sG[c * CCH + cp];
      }
    }
  }
  for (int o = 16; o > 0; o >>= 1) partial += __shfl_xor(partial, o, 32);
  if (lane == 0) sS[rn * 2 + rm] = partial;
  __syncthreads();

  // Per-parity 2x2 attention block (identical in all threads, tiny):
  // softmax over 4 cols: two real scores + two exact-zero scores (off3 cancels)
  const float off   = off_p[0];
  const float off2  = off2_p[0];
  const float inv_scale = 1.0f / sqrtf((float)(CCH * PP * PP));   // 1/sqrt(49152)

  float mtx[2][2];
#pragma unroll
  for (int r = 0; r < 2; ++r) {
    const float s0 = sS[r * 2 + 0] * inv_scale;
    const float s1 = sS[r * 2 + 1] * inv_scale;
    const float mx = fmaxf(fmaxf(s0, s1), 0.f);
    const float e0 = expf(s0 - mx);
    const float e1 = expf(s1 - mx);
    const float ez = expf(0.f - mx);
    const float denom = e0 + e1 + 2.f * ez;
    mtx[r][0] = e0 / denom + off2 + (r == 0 ? off : 0.f);
    mtx[r][1] = e1 / denom + off2 + (r == 1 ? off : 0.f);
  }

  if (tid == 0) {
    const float det = mtx[0][0] * mtx[1][1] - mtx[0][1] * mtx[1][0];
    ld_partial[gid] = logf(fabsf(det)) * (float)(PP * (PP / 2) * CCH);  // *24576
  }

  // Output pass: both patches of this parity, all pixels (coalesced along j).
  // pass-through where (j&1)==pi, else 2-term mix with the sibling patch.
  const int offP0 = patch_off(pi);
  const int offP1 = patch_off(pi + 2);
  float* ob = out + (size_t)b * CHW;
  const int total = 2 * CCH * PP * PP;   // 98304
  for (int idx = tid; idx < total; idx += 128) {
    const int p   = (idx >= CCH * PP * PP) ? 1 : 0;
    const int rem = idx - p * (CCH * PP * PP);
    const int c   = rem >> 10;
    const int pix = rem & 1023;
    const int i   = pix >> 5;
    const int j   = pix & 31;
    const int own_off   = (p ? offP1 : offP0) + i * WW + j;
    const int other_off = (p ? offP0 : offP1) + i * WW + j;
    const size_t cbase = (size_t)c * HW;
    const float own   = xb[cbase + own_off];
    const float other = xb[cbase + other_off];
    const float a  = mtx[p][p];
    const float bm = mtx[p][1 - p];
    const float val = ((j & 1) == pi) ? own : (a * own + bm * other);
    ob[cbase + own_off] = val;
  }
}

// ---------------------------------------------------------------------------
// Kernel 3: logdet_out[b] = logdet[b] + partial[2b] + partial[2b+1]
// ---------------------------------------------------------------------------
__global__ void k_logdet(const float* __restrict__ logdet,
                         const float* __restrict__ ld_partial,
                         float* __restrict__ out_ld) {
  int b = blockIdx.x * blockDim.x + threadIdx.x;
  if (b < BATCH) out_ld[b] = logdet[b] + ld_partial[2 * b] + ld_partial[2 * b + 1];
}

// ---------------------------------------------------------------------------
extern "C" void kernel_launch(void* const* d_in, const int* in_sizes, int n_in,
                              void* d_out, int out_size, void* d_ws, size_t ws_size,
                              hipStream_t stream) {
  (void)in_sizes; (void)n_in; (void)out_size; (void)ws_size;
  const float* x      = (const float*)d_in[0];
  const float* logdet = (const float*)d_in[1];
  const float* Wq1    = (const float*)d_in[2];
  const float* Wq2    = (const float*)d_in[3];
  const float* Wq3    = (const float*)d_in[4];
  const float* Wk1    = (const float*)d_in[5];
  const float* Wk2    = (const float*)d_in[6];
  const float* Wk3    = (const float*)d_in[7];
  const float* off    = (const float*)d_in[8];
  const float* off2   = (const float*)d_in[9];
  const float* off3   = (const float*)d_in[10];

  float* G   = (float*)d_ws;          // 2304 floats
  float* ldp = G + CCH * CCH;         // 256 floats

  k_build_G<<<(CCH * CCH + 255) / 256, 256, 0, stream>>>(Wq1, Wq2, Wq3, Wk1, Wk2, Wk3, G);
  k_main<<<BATCH * 2, 128, 0, stream>>>(x, G, off, off2, off3, (float*)d_out, ldp);
  k_logdet<<<1, 128, 0, stream>>>(logdet, ldp, (float*)d_out + OUT_IMG);
  (void)off3; // off3 cancels inside softmax (all 4 logits shift equally)
}